// EnergyLoc_18786186953534
// MI455X (gfx1250) — compile-verified
//
#include <hip/hip_runtime.h>
#include <hip/hip_bf16.h>

typedef float v2f __attribute__((ext_vector_type(2)));
typedef float v8f __attribute__((ext_vector_type(8)));

// One wave (32 lanes) per orbital, grid-stride over orbitals.
// e[n] = sum_t ampls[t] * ( v_t . (h1 x) + v_t^T (h2 : x x) v_t )
__global__ void __launch_bounds__(256)
energy_orbital_kernel(const float* __restrict__ h1,
                      const float* __restrict__ h2,
                      const float* __restrict__ v,      // [3, N, 8]
                      const float* __restrict__ ampls,  // [3]
                      const float* __restrict__ x,      // [N, 8]
                      float* __restrict__ e,            // [N] out
                      int n_orb) {
    const int lane   = threadIdx.x & 31;
    const int gwave  = (blockIdx.x * blockDim.x + threadIdx.x) >> 5;
    const int nwaves = (gridDim.x * blockDim.x) >> 5;

    const float a0 = ampls[0], a1 = ampls[1], a2 = ampls[2];

    const int hi  = lane >> 4;           // 0 for lanes 0-15, 1 for 16-31
    const int j0  = (4 * lane) & 63;     // fixed Q index base per lane
    const int c0  = j0 >> 3;             // Q row  (x[c0])
    const int d0  = j0 & 7;              // Q cols (x[d0..d0+3]), 0 or 4
    const int aL  = lane >> 3;           // Pf[lane]    = P[aL, bL]
    const int bL  = lane & 7;            // Pf[lane+32] = P[aL+4, bL]
    const int aE  = lane >> 2;           // h1 element pair (2l,2l+1): row
    const int bE  = (2 * lane) & 7;      // h1 element pair: cols bE, bE+1

    const size_t vstride = (size_t)n_orb * 8;

    for (int n = gwave; n < n_orb; n += nwaves) {
        const size_t xb  = (size_t)n * 8;
        const size_t h2b = (size_t)n * 4096;

        // ---- per-orbital small operands ----
        const float xreg = x[xb + (lane & 7)];          // lane holds x[lane%8]
        const float qx   = x[xb + c0];
        const float4 xd  = *(const float4*)(x + xb + d0);
        float4 qv;                                       // Qf[j0..j0+3]
        qv.x = qx * xd.x; qv.y = qx * xd.y; qv.z = qx * xd.z; qv.w = qx * xd.w;

        const float vr0 = v[0 * vstride + xb + (lane & 7)];   // raw v_t[lane%8]
        const float vr1 = v[1 * vstride + xb + (lane & 7)];
        const float vr2 = v[2 * vstride + xb + (lane & 7)];
        const float vw0 = a0 * vr0, vw1 = a1 * vr1, vw2 = a2 * vr2;
        const float pv  = vw0 + vw1 + vw2;               // lane holds pv[lane%8]

        // Pf[lane] and Pf[lane+32]: P[a,b] = sum_t a_t v_t[a] v_t[b]
        float pf_lo, pf_hi;
        {
            const float sb0 = __shfl(vr0, bL), sb1 = __shfl(vr1, bL), sb2 = __shfl(vr2, bL);
            float sa0 = __shfl(vw0, aL),     sa1 = __shfl(vw1, aL),     sa2 = __shfl(vw2, aL);
            pf_lo = fmaf(sa0, sb0, fmaf(sa1, sb1, sa2 * sb2));
            sa0 = __shfl(vw0, aL + 4); sa1 = __shfl(vw1, aL + 4); sa2 = __shfl(vw2, aL + 4);
            pf_hi = fmaf(sa0, sb0, fmaf(sa1, sb1, sa2 * sb2));
        }

        // ---- e1 part: sum_k h1flat[k] * pv[k>>3] * x[k&7], lane does k=2l,2l+1
        float acc0, acc1 = 0.0f;
        {
            const float2 h1e = *(const float2*)(h1 + (size_t)n * 64 + 2 * lane);
            const float xb0 = __shfl(xreg, bE);
            const float xb1 = __shfl(xreg, bE + 1);
            const float pva = __shfl(pv, aE);
            acc0 = pva * fmaf(h1e.y, xb1, h1e.x * xb0);
        }

        // warm next orbital's h2 tile
        if (n + nwaves < n_orb)
            __builtin_prefetch(h2 + h2b + (size_t)nwaves * 4096 + 4 * lane, 0, 0);

        // ---- main stream: 32 coalesced float4 chunks of h2 (16 KB/orbital)
        const float* __restrict__ hp = h2 + h2b + 4 * lane;
#pragma unroll
        for (int c = 0; c < 32; ++c) {
            const float4 h = *(const float4*)(hp + c * 128);
            float dot = h.x * qv.x;
            dot = fmaf(h.y, qv.y, dot);
            dot = fmaf(h.z, qv.z, dot);
            dot = fmaf(h.w, qv.w, dot);
            const float psrc = (c < 16) ? pf_lo : pf_hi;     // compile-time (unrolled)
            const float p = __shfl(psrc, (2 * c + hi) & 31); // Pf[2c + hi]
            if (c & 1) acc1 = fmaf(p, dot, acc1);
            else       acc0 = fmaf(p, dot, acc0);
        }

        // ---- exact f32 cross-lane reduction via WMMA ------------------------
        // A (16x4 f32): lane l<16 -> row l,  K0=acc0, K1=acc1
        //               lane l>=16 -> row l-16, K2=acc0, K3=acc1
        // B = all-ones (layout invariant)  =>  D[i][j] = rowsum_i for every j.
        v2f aF; aF[0] = acc0; aF[1] = acc1;
        v2f bF; bF[0] = 1.0f; bF[1] = 1.0f;
        v8f cF = {};
        v8f dF = __builtin_amdgcn_wmma_f32_16x16x4_f32(
            /*neg_a=*/false, aF, /*neg_b=*/false, bF,
            /*c_mod=*/(short)0, cF, /*reuse_a=*/false, /*reuse_b=*/false);
        float s = dF[0] + dF[1] + dF[2] + dF[3] + dF[4] + dF[5] + dF[6] + dF[7];
        // lanes 0-15 hold rows 0-7 summed, lanes 16-31 hold rows 8-15 summed
        const float tot = s + __shfl_xor(s, 16);
        if (lane == 0) e[n] = tot;
    }
}

// One wave per molecule: exclusive prefix of counts + segment sum + /denom.
__global__ void __launch_bounds__(128)
molecule_reduce_kernel(const float* __restrict__ e,
                       const int* __restrict__ counts,
                       const float* __restrict__ ampls,
                       float* __restrict__ out,
                       int n_mol) {
    const int lane = threadIdx.x & 31;
    const int m = blockIdx.x * (blockDim.x >> 5) + (threadIdx.x >> 5);
    if (m >= n_mol) return;                       // wave-uniform

    int pre = 0;
    for (int k = lane; k < m; k += 32) pre += counts[k];
#pragma unroll
    for (int off = 16; off; off >>= 1) pre += __shfl_xor(pre, off);

    const int cnt = counts[m];
    float s = 0.0f;
    for (int k = lane; k < cnt; k += 32) s += e[pre + k];
#pragma unroll
    for (int off = 16; off; off >>= 1) s += __shfl_xor(s, off);

    if (lane == 0) out[m] = s / (ampls[0] + ampls[1] + ampls[2]);
}

extern "C" void kernel_launch(void* const* d_in, const int* in_sizes, int n_in,
                              void* d_out, int out_size, void* d_ws, size_t ws_size,
                              hipStream_t stream) {
    const float* h1    = (const float*)d_in[0];   // [N,8,8]
    const float* h2    = (const float*)d_in[1];   // [N,8,8,8,8]
    const float* v     = (const float*)d_in[2];   // [3,N,8]
    const float* ampls = (const float*)d_in[3];   // [3]
    const float* x     = (const float*)d_in[4];   // [N,8]
    const int*   cnts  = (const int*)d_in[5];     // [M]
    float* out = (float*)d_out;

    const int n_orb = in_sizes[0] / 64;           // 60000
    const int n_mol = in_sizes[5];                // 3000

    float* e_ws = (float*)d_ws;                   // N floats of scratch

    // Enough waves for full occupancy; each wave streams ~4 orbitals (16 KB each).
    const int blocks1 = 2048, threads1 = 256;     // 16384 waves
    energy_orbital_kernel<<<blocks1, threads1, 0, stream>>>(
        h1, h2, v, ampls, x, e_ws, n_orb);

    const int wavesPerBlock = 4, threads2 = 32 * wavesPerBlock;
    const int blocks2 = (n_mol + wavesPerBlock - 1) / wavesPerBlock;
    molecule_reduce_kernel<<<blocks2, threads2, 0, stream>>>(
        e_ws, cnts, ampls, out, n_mol);
}